// HashConvNet_60017872994399
// MI455X (gfx1250) — compile-verified
//
#include <hip/hip_runtime.h>
#include <hip/hip_bf16.h>

typedef __attribute__((ext_vector_type(2))) float v2f;
typedef __attribute__((ext_vector_type(8))) float v8f;
typedef __attribute__((ext_vector_type(2))) int   v2i;

// ---------------- workspace layout (float offsets) ----------------
#define SZ_WM1   (96 * 76)          // K=75 padded to 76 (pad col = 0)
#define SZ_WM2   (192 * 864)
#define SZ_WM3   (384 * 1728)
#define OFF_WM1  0
#define OFF_WM2  (OFF_WM1 + SZ_WM1)
#define OFF_WM3  (OFF_WM2 + SZ_WM2)
#define OFF_XIMG (OFF_WM3 + SZ_WM3)         // padded copy of x_a
#define SZ_XIMG  (128 * 3 * 64 * 64)
#define OFF_ACT1 (OFF_XIMG + SZ_XIMG + 64)  // +64 tail pad on each buffer
#define SZ_ACT1  (128 * 96 * 30 * 30)
#define OFF_ACT2 (OFF_ACT1 + SZ_ACT1 + 64)
#define SZ_ACT2  (128 * 192 * 14 * 14)

// ---------------- output layout (float offsets) -------------------
#define OUT_RA   0                       // 128*10
#define OUT_RB   1280                    // 128*10
#define OUT_PRE1 2560                    // 128*192*14*14
#define OUT_PRE2 (2560 + 128 * 192 * 14 * 14)

// ==== fold +/-1 masks into weights (linearity) + copy x_a padded ====
__global__ void mask_weights_copy(const float* __restrict__ W1, const float* __restrict__ o1,
                                  const float* __restrict__ W2, const float* __restrict__ o2,
                                  const float* __restrict__ W3, const float* __restrict__ o3,
                                  const float* __restrict__ x,
                                  const int* __restrict__ timep, float* __restrict__ ws) {
  const int t = (*timep) & 63;  // time % 64
  const int totW = SZ_WM1 + SZ_WM2 + SZ_WM3;
  const int total = totW + SZ_XIMG;
  for (int i = blockIdx.x * blockDim.x + threadIdx.x; i < total;
       i += gridDim.x * blockDim.x) {
    if (i < SZ_WM1) {
      int m = i / 76, k = i % 76;
      ws[i] = (k < 75) ? W1[m * 75 + k] * o1[t * 3 + k / 25] : 0.f;
    } else if (i < SZ_WM1 + SZ_WM2) {
      int j = i - SZ_WM1;
      ws[i] = W2[j] * o2[t * 96 + (j % 864) / 9];
    } else if (i < totW) {
      int j = i - (SZ_WM1 + SZ_WM2);
      ws[i] = W3[j] * o3[t * 192 + (j % 1728) / 9];
    } else {
      int j = i - totW;
      ws[OFF_XIMG + j] = x[j];
    }
  }
}

// ============ layer 1: 5x5 conv (K=76 padded) + maxpool 2x2 + ReLU ============
// grid: (mg=3 [32ch], xtile=4, b*30+py), block = 64 (2 waves = 2 conv rows)
__global__ void __launch_bounds__(64)
conv1_pool(const float* __restrict__ ximg, const float* __restrict__ wm1,
           const float* __restrict__ b1, float* __restrict__ act1) {
  __shared__ __align__(16) int offt[76];
  __shared__ float tile[2][32][16];
  const int tid = threadIdx.x;
  for (int k = tid; k < 76; k += 64) {
    int cin = k / 25, r = k % 25;
    offt[k] = (k < 75) ? cin * 4096 + (r / 5) * 64 + (r % 5) : 0;
  }
  __syncthreads();

  const int mg = blockIdx.x;       // 32-channel group
  const int xtile = blockIdx.y;
  const int b  = blockIdx.z / 30;
  const int py = blockIdx.z % 30;
  const int lane = tid & 31, wave = tid >> 5;
  const int g = lane >> 4, ln = lane & 15;
  const int y = 2 * py + wave;                       // conv row 0..59
  const int hbase = y * 64 + 16 * xtile + ln;        // spatial base in image
  const float* ib = ximg + (size_t)b * 3 * 4096;
  const float* wrow = wm1 + (size_t)(mg * 32 + ln) * 76;

  v8f acc[2] = {};
  for (int k0 = 0; k0 < 76; k0 += 4) {
    const int ka = k0 + 2 * g;
    v2i off = *(const v2i*)&offt[ka];
    v2f B;
    B.x = ib[hbase + off.x];
    B.y = ib[hbase + off.y];
#pragma unroll
    for (int tt = 0; tt < 2; ++tt) {
      v2f A = *(const v2f*)(wrow + tt * 16 * 76 + ka);
      acc[tt] = __builtin_amdgcn_wmma_f32_16x16x4_f32(false, A, false, B, (short)0,
                                                      acc[tt], false, false);
    }
  }
#pragma unroll
  for (int tt = 0; tt < 2; ++tt)
#pragma unroll
    for (int r = 0; r < 8; ++r) tile[wave][tt * 16 + 8 * g + r][ln] = acc[tt][r];
  __syncthreads();

  for (int o = tid; o < 256; o += 64) {   // 32 ch x 8 pooled px
    int mm = o >> 3, px8 = o & 7;
    int px = xtile * 8 + px8;
    if (px < 30) {
      float v = fmaxf(fmaxf(tile[0][mm][2 * px8], tile[0][mm][2 * px8 + 1]),
                      fmaxf(tile[1][mm][2 * px8], tile[1][mm][2 * px8 + 1]));
      int ch = mg * 32 + mm;
      v = fmaxf(v + b1[ch], 0.f);
      act1[((b * 96 + ch) * 30 + py) * 30 + px] = v;
    }
  }
}

// ============ layer 2: 3x3 conv (K=864) + maxpool + pre1 + ReLU ============
// grid: (mg=3 [64ch], xtile=2, b*14+py), block = 64
__global__ void __launch_bounds__(64)
conv2_pool(const float* __restrict__ act1, const float* __restrict__ wm2,
           const float* __restrict__ b2, float* __restrict__ act2,
           float* __restrict__ pre1) {
  __shared__ __align__(16) int offt[864];
  __shared__ float tile[2][64][16];
  const int tid = threadIdx.x;
  for (int k = tid; k < 864; k += 64) {
    int r = k % 9;
    offt[k] = (k / 9) * 900 + (r / 3) * 30 + (r % 3);
  }
  __syncthreads();

  const int mg = blockIdx.x;
  const int xtile = blockIdx.y;
  const int b  = blockIdx.z / 14;
  const int py = blockIdx.z % 14;
  const int lane = tid & 31, wave = tid >> 5;
  const int g = lane >> 4, ln = lane & 15;
  const int y = 2 * py + wave;                       // 0..27
  const int hbase = y * 30 + 16 * xtile + ln;
  const float* ib = act1 + (size_t)b * 96 * 900;
  const float* wrow = wm2 + (size_t)(mg * 64 + ln) * 864;

  v8f acc[4] = {};
  for (int k0 = 0; k0 < 864; k0 += 4) {
    const int ka = k0 + 2 * g;
    v2i off = *(const v2i*)&offt[ka];
    v2f B;
    B.x = ib[hbase + off.x];
    B.y = ib[hbase + off.y];
#pragma unroll
    for (int tt = 0; tt < 4; ++tt) {
      v2f A = *(const v2f*)(wrow + tt * 16 * 864 + ka);
      acc[tt] = __builtin_amdgcn_wmma_f32_16x16x4_f32(false, A, false, B, (short)0,
                                                      acc[tt], false, false);
    }
  }
#pragma unroll
  for (int tt = 0; tt < 4; ++tt)
#pragma unroll
    for (int r = 0; r < 8; ++r) tile[wave][tt * 16 + 8 * g + r][ln] = acc[tt][r];
  __syncthreads();

  for (int o = tid; o < 512; o += 64) {   // 64 ch x 8 pooled px
    int mm = o >> 3, px8 = o & 7;
    int px = xtile * 8 + px8;
    if (px < 14) {
      float v = fmaxf(fmaxf(tile[0][mm][2 * px8], tile[0][mm][2 * px8 + 1]),
                      fmaxf(tile[1][mm][2 * px8], tile[1][mm][2 * px8 + 1]));
      int ch = mg * 64 + mm;
      v += b2[ch];
      int oidx = ((b * 192 + ch) * 14 + py) * 14 + px;
      pre1[oidx] = v;                 // pooled, pre-ReLU tensor output
      act2[oidx] = fmaxf(v, 0.f);
    }
  }
}

// ============ layer 3: 3x3 conv (K=1728), no pool, writes pre2 ============
// grid: (mg=6 [64ch], ntilepair=576), block = 64 (each wave one 16-wide N tile)
__global__ void __launch_bounds__(64)
conv3_kern(const float* __restrict__ act2, const float* __restrict__ wm3,
           const float* __restrict__ b3, float* __restrict__ pre2) {
  __shared__ __align__(16) int offt[1728];
  const int tid = threadIdx.x;
  for (int k = tid; k < 1728; k += 64) {
    int r = k % 9;
    offt[k] = (k / 9) * 196 + (r / 3) * 14 + (r % 3);
  }
  __syncthreads();

  const int mg = blockIdx.x;
  const int lane = tid & 31, wave = tid >> 5;
  const int g = lane >> 4, ln = lane & 15;
  const int n  = (blockIdx.y * 2 + wave) * 16 + ln;  // < 128*144
  const int b  = n / 144;
  const int rr = n % 144;
  const int oy = rr / 12, ox = rr % 12;
  const int hbase = oy * 14 + ox;
  const float* ib = act2 + (size_t)b * 192 * 196;
  const float* wrow = wm3 + (size_t)(mg * 64 + ln) * 1728;

  v8f acc[4] = {};
  for (int k0 = 0; k0 < 1728; k0 += 4) {
    const int ka = k0 + 2 * g;
    v2i off = *(const v2i*)&offt[ka];
    v2f B;
    B.x = ib[hbase + off.x];
    B.y = ib[hbase + off.y];
#pragma unroll
    for (int tt = 0; tt < 4; ++tt) {
      v2f A = *(const v2f*)(wrow + tt * 16 * 1728 + ka);
      acc[tt] = __builtin_amdgcn_wmma_f32_16x16x4_f32(false, A, false, B, (short)0,
                                                      acc[tt], false, false);
    }
  }
#pragma unroll
  for (int tt = 0; tt < 4; ++tt)
#pragma unroll
    for (int r = 0; r < 8; ++r) {
      int mm = mg * 64 + tt * 16 + 8 * g + r;
      pre2[((b * 384 + mm) * 12 + oy) * 12 + ox] = acc[tt][r] + b3[mm];
    }
}

// ============ FC: r_a = relu(pre2)*ofc @ Wfc^T + bfc ; r_b = bfc ============
__global__ void __launch_bounds__(256)
fc_kernel(const float* __restrict__ pre2, const float* __restrict__ Wfc,
          const float* __restrict__ bfc, const float* __restrict__ ofc,
          const int* __restrict__ timep, float* __restrict__ out) {
  const int b = blockIdx.x;
  const int tid = threadIdx.x;
  const int t = (*timep) & 63;
  const float* pb = pre2 + (size_t)b * 55296;
  const float* of = ofc + (size_t)t * 55296;
  float acc[10];
#pragma unroll
  for (int j = 0; j < 10; ++j) acc[j] = 0.f;
  for (int f = tid; f < 55296; f += 256) {
    float v = fmaxf(pb[f], 0.f) * of[f];
#pragma unroll
    for (int j = 0; j < 10; ++j) acc[j] += v * Wfc[j * 55296 + f];
  }
  __shared__ float red[256];
  for (int j = 0; j < 10; ++j) {
    __syncthreads();
    red[tid] = acc[j];
    __syncthreads();
    for (int s = 128; s > 0; s >>= 1) {
      if (tid < s) red[tid] += red[tid + s];
      __syncthreads();
    }
    if (tid == 0) out[OUT_RA + b * 10 + j] = red[0] + bfc[j];
  }
  if (tid < 10) out[OUT_RB + b * 10 + tid] = bfc[tid];  // b-path is identically zero
}

extern "C" void kernel_launch(void* const* d_in, const int* in_sizes, int n_in,
                              void* d_out, int out_size, void* d_ws, size_t ws_size,
                              hipStream_t stream) {
  (void)in_sizes; (void)n_in; (void)out_size; (void)ws_size;
  const float* x_a = (const float*)d_in[0];
  const float* W1  = (const float*)d_in[1];
  const float* b1  = (const float*)d_in[2];
  const float* o1  = (const float*)d_in[3];
  const float* W2  = (const float*)d_in[4];
  const float* b2  = (const float*)d_in[5];
  const float* o2  = (const float*)d_in[6];
  const float* W3  = (const float*)d_in[7];
  const float* b3  = (const float*)d_in[8];
  const float* o3  = (const float*)d_in[9];
  const float* Wfc = (const float*)d_in[10];
  const float* bfc = (const float*)d_in[11];
  const float* ofc = (const float*)d_in[12];
  const int* timep = (const int*)d_in[13];
  float* out = (float*)d_out;
  float* ws  = (float*)d_ws;

  float* wm1  = ws + OFF_WM1;
  float* wm2  = ws + OFF_WM2;
  float* wm3  = ws + OFF_WM3;
  float* ximg = ws + OFF_XIMG;
  float* act1 = ws + OFF_ACT1;
  float* act2 = ws + OFF_ACT2;
  float* pre1 = out + OUT_PRE1;
  float* pre2 = out + OUT_PRE2;

  mask_weights_copy<<<1024, 256, 0, stream>>>(W1, o1, W2, o2, W3, o3, x_a, timep, ws);
  conv1_pool<<<dim3(3, 4, 128 * 30), 64, 0, stream>>>(ximg, wm1, b1, act1);
  conv2_pool<<<dim3(3, 2, 128 * 14), 64, 0, stream>>>(act1, wm2, b2, act2, pre1);
  conv3_kern<<<dim3(6, 576, 1), 64, 0, stream>>>(act2, wm3, b3, pre2);
  fc_kernel<<<128, 256, 0, stream>>>(pre2, Wfc, bfc, ofc, timep, out);
}